// Net_34600256537163
// MI455X (gfx1250) — compile-verified
//
#include <hip/hip_runtime.h>
#include <math.h>

#define B_ROOT 8192
#define N_SUB 262144
#define F_ROOT 192
#define F_SUB 64
#define BAG 128
#define CLASSES 10
#define NEG_SLOPE 0.01f
#define LN_EPS 1e-5f
#define BAGS_PER_BLOCK 8

typedef float v2f __attribute__((ext_vector_type(2)));
typedef float v8f __attribute__((ext_vector_type(8)));

static __device__ __forceinline__ float leaky(float x) {
    return x > 0.f ? x : NEG_SLOPE * x;
}

// ---------------------------------------------------------------------------
// Kernel 1: exclusive prefix sum of lens -> off[B+1]
// ---------------------------------------------------------------------------
__global__ __launch_bounds__(1024) void k_offsets(const int* __restrict__ lens,
                                                  int* __restrict__ off) {
    __shared__ int part[1024];
    const int t = threadIdx.x;
    const int base = t * 8;
    int loc[8];
    int s = 0;
#pragma unroll
    for (int j = 0; j < 8; ++j) { loc[j] = lens[base + j]; s += loc[j]; }
    part[t] = s;
    __syncthreads();
    for (int d = 1; d < 1024; d <<= 1) {
        int v = (t >= d) ? part[t - d] : 0;
        __syncthreads();
        part[t] += v;
        __syncthreads();
    }
    int run = part[t] - s;  // exclusive prefix
#pragma unroll
    for (int j = 0; j < 8; ++j) { off[base + j] = run; run += loc[j]; }
    if (t == 1023) off[B_ROOT] = run;
}

// ---------------------------------------------------------------------------
// Kernel 2: fused sub pipeline, BAGS_PER_BLOCK bags per block so the 34 KB
// weight tile is loaded into LDS once per 8 bags (cuts L2 weight traffic 8x).
//   x_raw = leaky(concat(sub_feats, sub_mask) @ W_sub_in + b)   (WMMA f32)
//   outsub_partial[i] = x_raw[i,:] . W_out_sub[128:256]
//   sub_norm[b] = LayerNorm(segment_mean(x_raw))
// x_raw never touches global memory.
// ---------------------------------------------------------------------------
__global__ __launch_bounds__(256) void k_sub(
    const float* __restrict__ sub_feats, const float* __restrict__ sub_mask,
    const int* __restrict__ off,
    const float* __restrict__ W_sub_in, const float* __restrict__ b_sub_in,
    const float* __restrict__ g_sub, const float* __restrict__ be_sub,
    const float* __restrict__ W_out_sub,     // [256,1] flat
    float* __restrict__ sub_norm,            // [B,128] ws
    float* __restrict__ outsub)              // [N] slice of d_out
{
    __shared__ float Bs[68][128];   // W_sub_in, K padded 65->68
    __shared__ float As[16][68];    // input tile
    __shared__ float acc[128];      // segment sum
    __shared__ float rowdot[16];    // per-row sub-dot
    __shared__ float red1[128];
    __shared__ float red2[128];

    const int tid  = threadIdx.x;
    const int w    = tid >> 5;
    const int lane = tid & 31;
    const int half = lane >> 4;
    const int l    = lane & 15;
    const int col  = (w << 4) + l;  // 0..127

    // load weights once per block
    for (int idx = tid; idx < 68 * 128; idx += 256) {
        int r = idx >> 7, c = idx & 127;
        Bs[r][c] = (r < 65) ? W_sub_in[r * 128 + c] : 0.f;
    }
    const float bias = b_sub_in[col];
    const float w1   = W_out_sub[128 + col];

    for (int bb = 0; bb < BAGS_PER_BLOCK; ++bb) {
        const int bag   = blockIdx.x * BAGS_PER_BLOCK + bb;
        const int start = off[bag];
        const int len   = off[bag + 1] - start;

        if (tid < 128) acc[tid] = 0.f;
        __syncthreads();   // guards Bs (first bag), acc reset, prior-bag LN reads

        const int nchunk = (len + 15) >> 4;
        for (int ch = 0; ch < nchunk; ++ch) {
            // stage 16x68 A tile (rows past len and K-pad cols are zero)
            for (int idx = tid; idx < 16 * 68; idx += 256) {
                int r = idx / 68, c = idx - r * 68;
                int gi = (ch << 4) + r;
                float v = 0.f;
                if (gi < len) {
                    int g = start + gi;
                    if (c < 64)       v = sub_feats[(size_t)g * 64 + c];
                    else if (c == 64) v = sub_mask[g];
                }
                As[r][c] = v;
            }
            if (tid < 16) {
                rowdot[tid] = 0.f;
                // prefetch next chunk's rows (global_prefetch_b8)
                int gi = ((ch + 1) << 4) + tid;
                if (gi < len)
                    __builtin_prefetch(&sub_feats[(size_t)(start + gi) * 64], 0, 0);
            }
            __syncthreads();

            v8f c8 = {};
#pragma unroll
            for (int ks = 0; ks < 17; ++ks) {
                v2f a, bm;
                a.x  = As[l][ks * 4 + 2 * half];
                a.y  = As[l][ks * 4 + 2 * half + 1];
                bm.x = Bs[ks * 4 + 2 * half][col];
                bm.y = Bs[ks * 4 + 2 * half + 1][col];
                c8 = __builtin_amdgcn_wmma_f32_16x16x4_f32(
                    false, a, false, bm, (short)0, c8, false, false);
            }

            // epilogue: bias + leaky; per-lane segment-sum; per-row dot with W1
            float vs = 0.f;
#pragma unroll
            for (int r = 0; r < 8; ++r) {
                int m = (half << 3) + r;                 // local row 0..15
                bool valid = ((ch << 4) + m) < len;
                float v = leaky(c8[r] + bias);
                float p = 0.f;
                if (valid) { vs += v; p = v * w1; }
#pragma unroll
                for (int o = 8; o >= 1; o >>= 1) p += __shfl_xor(p, o, 32);
                if (l == 0) atomicAdd(&rowdot[m], p);
            }
            atomicAdd(&acc[col], vs);   // one ds_add_f32 per lane per chunk
            __syncthreads();
            if (tid < 16) {
                int gi = (ch << 4) + tid;
                if (gi < len) outsub[start + gi] = rowdot[tid];
            }
            __syncthreads();
        }

        // segment mean + LayerNorm over 128 features
        if (tid < 128) {
            float inv = 1.f / (float)(len > 0 ? len : 1);
            float a = acc[tid] * inv;
            acc[tid] = a;
            red1[tid] = a;
            red2[tid] = a * a;
        }
        __syncthreads();
        for (int d = 64; d >= 1; d >>= 1) {
            if (tid < d) { red1[tid] += red1[tid + d]; red2[tid] += red2[tid + d]; }
            __syncthreads();
        }
        if (tid < 128) {
            float mu  = red1[0] * (1.f / 128.f);
            float var = red2[0] * (1.f / 128.f) - mu * mu;
            float rs  = rsqrtf(var + LN_EPS);
            sub_norm[(size_t)bag * 128 + tid] =
                (acc[tid] - mu) * rs * g_sub[tid] + be_sub[tid];
        }
        // next bag's acc-reset barrier separates these reads from red1/red2 reuse
    }
}

// ---------------------------------------------------------------------------
// Kernel 3: root GEMM (WMMA f32) + LayerNorm + all heads + r_dot.
// One block per 16 rows; K = 194 padded to 196, 7 chunks of 28.
// ---------------------------------------------------------------------------
__global__ __launch_bounds__(256) void k_root(
    const float* __restrict__ root_feats, const float* __restrict__ root_mask,
    const float* __restrict__ sub_norm,
    const float* __restrict__ W_root_in, const float* __restrict__ b_root_in,
    const float* __restrict__ g_root, const float* __restrict__ be_root,
    const float* __restrict__ W_v, const float* __restrict__ b_v,
    const float* __restrict__ W_cls_p, const float* __restrict__ b_cls_p,
    const float* __restrict__ W_cls_a, const float* __restrict__ b_cls_a,
    const float* __restrict__ W_out_root, const float* __restrict__ b_out_root,
    const float* __restrict__ W_out_sub,
    float* __restrict__ o_v, float* __restrict__ o_cls_p,
    float* __restrict__ of_x, float* __restrict__ r_dot)
{
    __shared__ float Bw[28][128];
    __shared__ float Ax[16][28];
    __shared__ float xL[16][128];

    const int tid  = threadIdx.x;
    const int rowBase = blockIdx.x << 4;
    const int w    = tid >> 5;
    const int lane = tid & 31;
    const int half = lane >> 4;
    const int l    = lane & 15;
    const int col  = (w << 4) + l;

    v8f c8 = {};
    for (int kc = 0; kc < 7; ++kc) {
        const int kbase = kc * 28;
        for (int idx = tid; idx < 28 * 128; idx += 256) {
            int r = idx >> 7, c = idx & 127;
            int kr = kbase + r;
            Bw[r][c] = (kr < 194) ? W_root_in[kr * 128 + c] : 0.f;
        }
        for (int idx = tid; idx < 16 * 28; idx += 256) {
            int r = idx / 28, c = idx - r * 28;
            int gc = kbase + c;
            int row = rowBase + r;
            float v = 0.f;
            if (gc < 64)       v = root_feats[(size_t)row * F_ROOT + gc];
            else if (gc < 192) v = sub_norm[(size_t)row * 128 + (gc - 64)];
            else if (gc < 194) v = root_mask[row * 2 + (gc - 192)];
            Ax[r][c] = v;
        }
        __syncthreads();
#pragma unroll
        for (int ks = 0; ks < 7; ++ks) {
            v2f a, bm;
            a.x  = Ax[l][ks * 4 + 2 * half];
            a.y  = Ax[l][ks * 4 + 2 * half + 1];
            bm.x = Bw[ks * 4 + 2 * half][col];
            bm.y = Bw[ks * 4 + 2 * half + 1][col];
            c8 = __builtin_amdgcn_wmma_f32_16x16x4_f32(
                false, a, false, bm, (short)0, c8, false, false);
        }
        __syncthreads();
    }

    // bias + LeakyReLU -> xL
    const float bias = b_root_in[col];
#pragma unroll
    for (int r = 0; r < 8; ++r) {
        int m = (half << 3) + r;
        xL[m][col] = leaky(c8[r] + bias);
    }
    __syncthreads();

    // LayerNorm per row: 16-thread groups (lanes 0-15 / 16-31 of each wave)
    {
        const int row = tid >> 4;
        const int lg  = tid & 15;
        float s = 0.f, s2 = 0.f;
#pragma unroll
        for (int j = 0; j < 8; ++j) {
            float x = xL[row][lg + 16 * j];
            s += x; s2 += x * x;
        }
#pragma unroll
        for (int o = 8; o >= 1; o >>= 1) {
            s  += __shfl_xor(s, o, 32);
            s2 += __shfl_xor(s2, o, 32);
        }
        float mu  = s * (1.f / 128.f);
        float var = s2 * (1.f / 128.f) - mu * mu;
        float rs  = rsqrtf(var + LN_EPS);
#pragma unroll
        for (int j = 0; j < 8; ++j) {
            int c = lg + 16 * j;
            float x = xL[row][c];
            xL[row][c] = (x - mu) * rs * g_root[c] + be_root[c];
        }
    }
    __syncthreads();

    // heads
    {
        const int row  = tid >> 4;
        const int lg   = tid & 15;
        const int grow = rowBase + row;
        float xv[8];
#pragma unroll
        for (int j = 0; j < 8; ++j) xv[j] = xL[row][lg + 16 * j];

        float dv = 0.f, da = 0.f, d0 = 0.f, d1 = 0.f, dr = 0.f;
#pragma unroll
        for (int j = 0; j < 8; ++j) {
            int c = lg + 16 * j;
            dv += xv[j] * W_v[c];
            da += xv[j] * W_cls_a[c];
            d0 += xv[j] * W_out_root[c * 2 + 0];
            d1 += xv[j] * W_out_root[c * 2 + 1];
            dr += xv[j] * W_out_sub[c];
        }
#pragma unroll
        for (int o = 8; o >= 1; o >>= 1) {
            dv += __shfl_xor(dv, o, 32);
            da += __shfl_xor(da, o, 32);
            d0 += __shfl_xor(d0, o, 32);
            d1 += __shfl_xor(d1, o, 32);
            dr += __shfl_xor(dr, o, 32);
        }
        if (lg == 0) {
            o_v[grow] = dv + b_v[0];
            of_x[grow * 3 + 0] = da + b_cls_a[0];
            float m0 = root_mask[grow * 2 + 0];
            float m1 = root_mask[grow * 2 + 1];
            of_x[grow * 3 + 1] = (m0 >= 1.f) ? -__builtin_inff() : (d0 + b_out_root[0]);
            of_x[grow * 3 + 2] = (m1 >= 1.f) ? -__builtin_inff() : (d1 + b_out_root[1]);
            r_dot[grow] = dr;
        }
        for (int cc = 0; cc < CLASSES; ++cc) {
            float dp = 0.f;
#pragma unroll
            for (int j = 0; j < 8; ++j) {
                int c = lg + 16 * j;
                dp += xv[j] * W_cls_p[c * CLASSES + cc];
            }
#pragma unroll
            for (int o = 8; o >= 1; o >>= 1) dp += __shfl_xor(dp, o, 32);
            if (lg == 0) o_cls_p[grow * CLASSES + cc] = dp + b_cls_p[cc];
        }
    }
}

// ---------------------------------------------------------------------------
// Kernel 4: sub_out[i] = mask ? -inf : partial[i] + r_dot[bag] + b_out_sub
// ---------------------------------------------------------------------------
__global__ __launch_bounds__(128) void k_fin(
    const int* __restrict__ off, const float* __restrict__ sub_mask,
    const float* __restrict__ r_dot, const float* __restrict__ b_out_sub,
    float* __restrict__ outsub)
{
    const int bag = blockIdx.x;
    const int start = off[bag];
    const int len   = off[bag + 1] - start;
    const float add = r_dot[bag] + b_out_sub[0];
    for (int i = threadIdx.x; i < len; i += 128) {
        int g = start + i;
        float v = outsub[g] + add;
        outsub[g] = (sub_mask[g] >= 1.f) ? -__builtin_inff() : v;
    }
}

// ---------------------------------------------------------------------------
extern "C" void kernel_launch(void* const* d_in, const int* in_sizes, int n_in,
                              void* d_out, int out_size, void* d_ws, size_t ws_size,
                              hipStream_t stream)
{
    const float* root_feats = (const float*)d_in[0];
    const float* root_mask  = (const float*)d_in[1];
    const float* sub_feats  = (const float*)d_in[2];
    const float* sub_mask   = (const float*)d_in[3];
    const int*   lens       = (const int*)d_in[4];
    const float* W_sub_in   = (const float*)d_in[5];
    const float* b_sub_in   = (const float*)d_in[6];
    const float* g_sub      = (const float*)d_in[7];
    const float* be_sub     = (const float*)d_in[8];
    const float* W_root_in  = (const float*)d_in[9];
    const float* b_root_in  = (const float*)d_in[10];
    const float* g_root     = (const float*)d_in[11];
    const float* be_root    = (const float*)d_in[12];
    const float* W_v        = (const float*)d_in[13];
    const float* b_v        = (const float*)d_in[14];
    const float* W_cls_p    = (const float*)d_in[15];
    const float* b_cls_p    = (const float*)d_in[16];
    const float* W_cls_a    = (const float*)d_in[17];
    const float* b_cls_a    = (const float*)d_in[18];
    const float* W_out_root = (const float*)d_in[19];
    const float* b_out_root = (const float*)d_in[20];
    const float* W_out_sub  = (const float*)d_in[21];
    const float* b_out_sub  = (const float*)d_in[22];

    float* o_v    = (float*)d_out;                 // [B]
    float* o_cls  = o_v + B_ROOT;                  // [B,10]
    float* of_x   = o_cls + B_ROOT * CLASSES;      // [B,3]
    float* outsub = of_x + B_ROOT * 3;             // [N]

    char* ws = (char*)d_ws;
    int*   off      = (int*)ws;                                      // (B+1) ints
    float* sub_norm = (float*)(ws + 33024);                          // [B,128]
    float* r_dot    = (float*)(ws + 33024 + (size_t)B_ROOT * 128 * 4);

    k_offsets<<<1, 1024, 0, stream>>>(lens, off);
    k_sub<<<B_ROOT / BAGS_PER_BLOCK, 256, 0, stream>>>(
        sub_feats, sub_mask, off, W_sub_in, b_sub_in, g_sub, be_sub,
        W_out_sub, sub_norm, outsub);
    k_root<<<B_ROOT / 16, 256, 0, stream>>>(root_feats, root_mask, sub_norm,
                                            W_root_in, b_root_in, g_root, be_root,
                                            W_v, b_v, W_cls_p, b_cls_p,
                                            W_cls_a, b_cls_a,
                                            W_out_root, b_out_root, W_out_sub,
                                            o_v, o_cls, of_x, r_dot);
    k_fin<<<B_ROOT, 128, 0, stream>>>(off, sub_mask, r_dot, b_out_sub, outsub);
}